// FocalLoss_78340203479413
// MI455X (gfx1250) — compile-verified
//
#include <hip/hip_runtime.h>

// Focal loss (ALPHA=0.25, GAMMA=2, IGNORE=255) over [4,19,512,512] fp32 logits
// + [4,512,512] int32 labels -> scalar mean. Bandwidth-bound (~84 MB, ~3.6 us
// at 23.3 TB/s). Kernel 1 streams logits with 128-bit loads, reusing each
// label quad across all 19 channels; kernel 2 reduces the 1024 block partials
// on the matrix pipe with V_WMMA_F32_16X16X4_F32 (B = ones => row sums).

#define NB 4
#define NC 19
#define HH 512
#define WW 512
#define HW (HH * WW)          // 262144
#define NPIX (NB * HW)        // 1048576
#define QUADS (NPIX / 4)      // 262144
#define NBLOCKS 1024
#define NTHREADS 256
#define ALPHA_F 0.25f
#define IGNORE_INDEX 255
#define INV_TOTAL (1.0f / (float)(NB * NC * HW))   // LOSS_WEIGHT == 1.0

typedef __attribute__((ext_vector_type(2))) float v2f;
typedef __attribute__((ext_vector_type(8))) float v8f;

__device__ __forceinline__ float focal_elem(float xv, int lab, int c) {
    const bool valid = (lab >= 0) && (lab != IGNORE_INDEX);
    const bool hot   = valid && (lab == c);
    // class weight from RAW logits (matches reference): a*(1-x)^2 if onehot else a*x^2
    const float d = hot ? (1.0f - xv) : xv;
    const float w = ALPHA_F * d * d;
    // BCE with logits, stable form
    const float bce = fmaxf(xv, 0.0f) - (hot ? xv : 0.0f)
                    + __logf(1.0f + __expf(-fabsf(xv)));
    return valid ? (bce * w) : 0.0f;
}

__global__ void __launch_bounds__(NTHREADS)
focal_partial_kernel(const float* __restrict__ x,
                     const int* __restrict__ lab,
                     float* __restrict__ partial) {
    float acc = 0.0f;
    for (int q = blockIdx.x * NTHREADS + threadIdx.x; q < QUADS;
         q += gridDim.x * NTHREADS) {
        const int n   = q >> 16;              // q / (HW/4), HW/4 = 65536
        const int rem = q & 0xFFFF;           // q % (HW/4)
        const int pix = n * HW + rem * 4;
        const int4 l4 = *reinterpret_cast<const int4*>(lab + pix);
        const float* base = x + (n * NC) * HW + rem * 4;
#pragma unroll
        for (int c = 0; c < NC; ++c) {
            const float4 v = *reinterpret_cast<const float4*>(base + c * HW);
            acc += focal_elem(v.x, l4.x, c);
            acc += focal_elem(v.y, l4.y, c);
            acc += focal_elem(v.z, l4.z, c);
            acc += focal_elem(v.w, l4.w, c);
        }
    }
    // wave32 butterfly reduction
#pragma unroll
    for (int off = 16; off > 0; off >>= 1)
        acc += __shfl_xor(acc, off, 32);
    __shared__ float smem[NTHREADS / 32];
    const int lane = threadIdx.x & 31;
    const int wid  = threadIdx.x >> 5;
    if (lane == 0) smem[wid] = acc;
    __syncthreads();
    if (threadIdx.x == 0) {
        float s = 0.0f;
#pragma unroll
        for (int i = 0; i < NTHREADS / 32; ++i) s += smem[i];
        partial[blockIdx.x] = s;
    }
}

// One wave. 16 chained f32 WMMAs: A holds 64 partials per step, B = ones,
// so D accumulates row sums; every column of D is identical. Final total =
// per-lane sum of the 8 accumulator VGPRs + one xor-16 shuffle.
__global__ void __launch_bounds__(32)
focal_final_kernel(const float* __restrict__ partial, float* __restrict__ out) {
    const int lane = threadIdx.x;          // 0..31, EXEC all ones for WMMA
    const v2f bones = {1.0f, 1.0f};
    v8f c = {};
#pragma unroll
    for (int i = 0; i < NBLOCKS / 64; ++i) {
        v2f a;
        a.x = partial[i * 64 + lane];
        a.y = partial[i * 64 + 32 + lane];
        // 8 args: (neg_a, A, neg_b, B, c_mod, C, reuse_a, reuse_b)
        c = __builtin_amdgcn_wmma_f32_16x16x4_f32(
                false, a, false, bones, (short)0, c, false, false);
    }
    float s = c[0] + c[1] + c[2] + c[3] + c[4] + c[5] + c[6] + c[7];
    s += __shfl_xor(s, 16, 32);            // fold M=0..7 half with M=8..15 half
    if (lane == 0) out[0] = s * INV_TOTAL;
}

extern "C" void kernel_launch(void* const* d_in, const int* in_sizes, int n_in,
                              void* d_out, int out_size, void* d_ws, size_t ws_size,
                              hipStream_t stream) {
    const float* x  = (const float*)d_in[0];   // cls_score [4,19,512,512] fp32
    const int* lab  = (const int*)d_in[1];     // label [4,512,512] int32
    float* partial  = (float*)d_ws;            // 1024 floats of scratch
    float* outp     = (float*)d_out;

    focal_partial_kernel<<<NBLOCKS, NTHREADS, 0, stream>>>(x, lab, partial);
    focal_final_kernel<<<1, 32, 0, stream>>>(partial, outp);
}